// PatchedGemmaAttention_25469156065900
// MI455X (gfx1250) — compile-verified
//
#include <hip/hip_runtime.h>
#include <hip/hip_bf16.h>
#include <math.h>

// ---------------------------------------------------------------------------
// Gemma-style paged attention decode on gfx1250 (MI455X), fp32 WMMA 16x16x4,
// with double-buffered Tensor Data Mover (TDM) staging of paged KV tiles.
//   B=16, BLOCK=64, MAX_BLOCKS=32, NUM_BLOCKS=512, MAX_KV=2048
//   HQ=16, HK=8 (G=2), D=256, HIDDEN=3072, MAX_Q=4
// ---------------------------------------------------------------------------

#define HQ_ 16
#define HK_ 8
#define D_ 256
#define HIDDEN_ 3072
#define MAXBLK_ 32
#define BLK_ 64

typedef float v2f __attribute__((ext_vector_type(2)));
typedef float v8f __attribute__((ext_vector_type(8)));
typedef unsigned int u32x4 __attribute__((ext_vector_type(4)));
typedef int i32x4 __attribute__((ext_vector_type(4)));
typedef int i32x8 __attribute__((ext_vector_type(8)));

// D = A(16x4 f32) * B(4x16 f32) + C(16x16 f32)
// args: (neg_a, A, neg_b, B, c_mod, C, reuse_a, reuse_b)
__device__ __forceinline__ v8f wmma4(v2f a, v2f b, v8f c) {
  return __builtin_amdgcn_wmma_f32_16x16x4_f32(false, a, false, b, (short)0, c,
                                               false, false);
}

// ---------------------------------------------------------------------------
// TDM: DMA a 2D tile of 16 rows x 256 fp32 (row stride 2048 fp32) from global
// memory into LDS at byte offset lds_off, inserting 4 DWORDs of padding after
// every 256 DWORDs so rows land at the 260-float LDS pitch used by the WMMA
// fragment gathers. Issued once per wave; completion tracked with TENSORcnt.
//
// D# layout per CDNA5 ISA ch.10.8:
//  group0: [1:0]=count=1 | [63:32]=lds_addr | [120:64]=global_addr | type=2
//  group1: data_size=4B | pad_enable, interval=256dw, amount=4dw |
//          tensor_dim0=256, tensor_dim1=16, tile_dim0=256, tile_dim1=16,
//          tensor_dim0_stride=2048
// ---------------------------------------------------------------------------
__device__ __forceinline__ void tdm_load_tile_2d(unsigned int lds_off,
                                                 const void* gptr) {
  unsigned long long ga = (unsigned long long)(uintptr_t)gptr;
  u32x4 g0;
  g0[0] = 1u;  // count=1, user descriptor, no gather
  g0[1] = lds_off;
  g0[2] = (unsigned int)(ga & 0xFFFFFFFFu);
  g0[3] = (unsigned int)((ga >> 32) & 0x01FFFFFFu) | (2u << 30);  // type=2
  i32x8 g1;
  g1[0] = (int)((2u << 16)      // data_size = 4 bytes
                | (1u << 20)    // pad_enable
                | (7u << 22)    // pad_interval: 256 DWORDs
                | (3u << 25));  // pad_amount: 4 DWORDs
  g1[1] = (int)(256u << 16);    // tensor_dim0[15:0] = 256
  g1[2] = (int)(16u << 16);     // tensor_dim1[15:0] = 16
  g1[3] = (int)(256u << 16);    // tile_dim0 = 256
  g1[4] = (int)16;              // tile_dim1 = 16, tile_dim2 = 0
  g1[5] = (int)2048;            // tensor_dim0_stride = HK*D = 2048 elements
  g1[6] = 0;
  g1[7] = 0;
  i32x4 z = {0, 0, 0, 0};
#if defined(__clang_major__) && (__clang_major__ >= 23)
  i32x8 z8 = {0, 0, 0, 0, 0, 0, 0, 0};
  __builtin_amdgcn_tensor_load_to_lds(g0, g1, z, z, z8, 0);
#else
  __builtin_amdgcn_tensor_load_to_lds(g0, g1, z, z, 0);
#endif
}

// ---------------------------------------------------------------------------
// Generic fp32 GEMM: C[M,N] = A[M,K] @ W[K,N]  (row-major, N,K multiples of 32)
// block = 256 threads = 8 waves; tile = 16(M) x 128(N); wave w -> 16-col slab.
// A chunk (16x32) staged in LDS; W streamed coalesced (lanes 0..15 contiguous).
// ---------------------------------------------------------------------------
__global__ void gemm_f32_wmma(const float* __restrict__ A,
                              const float* __restrict__ W,
                              float* __restrict__ C, int M, int N, int K) {
  __shared__ float As[16][36];  // padded: 36*l mod 64 distinct for l=0..15
  const int tid = threadIdx.x;
  const int wave = tid >> 5;
  const int lane = tid & 31;
  const int lo = lane >> 4;  // half-wave: 0 or 1
  const int ln = lane & 15;
  const int m0 = blockIdx.y * 16;
  const int nl = blockIdx.x * 128 + wave * 16 + ln;  // this lane's column

  v8f acc = {0.f, 0.f, 0.f, 0.f, 0.f, 0.f, 0.f, 0.f};

  for (int k0 = 0; k0 < K; k0 += 32) {
    // cooperative stage of A[m0:m0+16, k0:k0+32]
    for (int e = tid; e < 16 * 32; e += 256) {
      int r = e >> 5, kk = e & 31;
      As[r][kk] = (m0 + r < M) ? A[(size_t)(m0 + r) * K + k0 + kk] : 0.0f;
    }
    __syncthreads();
    if (k0 + 32 < K)  // emits global_prefetch_b8 on gfx1250
      __builtin_prefetch(&W[(size_t)(k0 + 32) * N + nl], 0, 1);
#pragma unroll
    for (int kc = 0; kc < 32; kc += 4) {
      v2f a, b;
      a.x = As[ln][kc + 2 * lo];
      a.y = As[ln][kc + 2 * lo + 1];
      const float* wp = &W[(size_t)(k0 + kc + 2 * lo) * N + nl];
      b.x = wp[0];
      b.y = wp[N];
      acc = wmma4(a, b, acc);
    }
    __syncthreads();
  }
#pragma unroll
  for (int i = 0; i < 8; ++i) {
    int row = m0 + i + 8 * lo;  // lanes 16..31 hold rows M=8..15 of the tile
    if (row < M) C[(size_t)row * N + nl] = acc[i];
  }
}

// ---------------------------------------------------------------------------
// Neox rotate-half RoPE, in place on workspace q [T,HQ,D] and k [T,HK,D].
// ---------------------------------------------------------------------------
__global__ void rope_kernel(float* __restrict__ q_ws, float* __restrict__ k_ws,
                            const int* __restrict__ pos_ids,
                            const float* __restrict__ inv_freq, int T) {
  int idx = blockIdx.x * blockDim.x + threadIdx.x;
  const int nq = T * HQ_ * (D_ / 2);
  const int nk = T * HK_ * (D_ / 2);
  float* p;
  int t, i;
  if (idx < nq) {
    t = idx / (HQ_ * 128);
    int rem = idx % (HQ_ * 128);
    p = q_ws + (size_t)t * (HQ_ * D_) + (rem / 128) * D_;
    i = rem & 127;
  } else if (idx < nq + nk) {
    int r = idx - nq;
    t = r / (HK_ * 128);
    int rem = r % (HK_ * 128);
    p = k_ws + (size_t)t * (HK_ * D_) + (rem / 128) * D_;
    i = rem & 127;
  } else {
    return;
  }
  float f = (float)pos_ids[t] * inv_freq[i];
  float c = cosf(f), s = sinf(f);
  float x1 = p[i], x2 = p[i + 128];
  p[i] = x1 * c - x2 * s;
  p[i + 128] = x2 * c + x1 * s;
}

// ---------------------------------------------------------------------------
// Paged attention, flash-style online softmax with a double-buffered TDM
// pipeline. grid = (HK, B); block = 128 threads = 4 waves.
//
// Q panel rows r = g*4 + j (8 used, 8 zero pad), pre-scaled by D^-0.5.
// A 16-token aligned tile fully inside the history region lives inside ONE
// cache block (16 | 64), so wave 0 DMAs it with TENSOR_LOAD_TO_LDS into
// ping-pong LDS buffers: tile i+1's DMA is issued BEFORE waiting for tile i
// (tensor ops complete in order per wave, so s_wait_tensorcnt(2) releases
// tile i while tile i+1 streams in behind the WMMA work). Boundary/tail tiles
// (straddling hist, at most 2) fall back to per-lane float4 staging merging
// cache tokens with freshly projected k/v (inputs never mutated -- equivalent
// to the reference's cache scatter).
// Wave 0: QK^T scores via WMMA; threads 0..15: online softmax; all 4 waves:
// alpha-rescale accumulators + P@V WMMA over their 64-column slab of D.
// ---------------------------------------------------------------------------
__global__ void paged_attn_wmma(const float* __restrict__ q_ws,
                                const float* __restrict__ k_ws,
                                const float* __restrict__ v_ws,
                                const float* __restrict__ k_cache,
                                const float* __restrict__ v_cache,
                                const int* __restrict__ q_start_loc,
                                const int* __restrict__ q_seqlens,
                                const int* __restrict__ kv_seqlens,
                                const int* __restrict__ block_offsets,
                                float* __restrict__ o_ws, int T) {
  const int h = blockIdx.x;  // kv head
  const int b = blockIdx.y;  // sequence
  const int tid = threadIdx.x;
  const int wave = tid >> 5;
  const int lane = tid & 31;
  const int lo = lane >> 4;
  const int ln = lane & 15;

  __shared__ float Qs[16][260];
  __shared__ float Kd[2][16][260];  // ping-pong K tiles (TDM destinations)
  __shared__ float Vd[2][16][260];  // ping-pong V tiles
  __shared__ float Ps[16][17];
  __shared__ float mS[16], lS[16], aS[16];

  const int q_len = q_seqlens[b];
  const int kv_len = kv_seqlens[b];
  const int hist = kv_len - q_len;
  const int q_start = q_start_loc[b];

  unsigned int ldsKb[2], ldsVb[2];
  ldsKb[0] = (unsigned int)(uintptr_t)&Kd[0][0][0];
  ldsKb[1] = (unsigned int)(uintptr_t)&Kd[1][0][0];
  ldsVb[0] = (unsigned int)(uintptr_t)&Vd[0][0][0];
  ldsVb[1] = (unsigned int)(uintptr_t)&Vd[1][0][0];

  const int nTiles = (kv_len + 15) >> 4;
  const int pureTiles = hist >> 4;  // tiles fully inside the history region

  // issue TDM K+V loads for pure-history tile ti2 into buffer (ti2 & 1)
  auto issue_tile = [&](int ti2) {
    int ss = ti2 * 16;
    int blk = block_offsets[b * MAXBLK_ + (ss >> 6)];
    size_t base = (((size_t)blk * BLK_ + (ss & 63)) * HK_ + h) * D_;
    int pb = ti2 & 1;
    tdm_load_tile_2d(ldsKb[pb], &k_cache[base]);
    tdm_load_tile_2d(ldsVb[pb], &v_cache[base]);
  };

  // prologue: start tile 0's DMA; it overlaps the Q panel staging below
  if (wave == 0 && pureTiles > 0) issue_tile(0);

  // ---- load Q panel (scaled by 1/sqrt(D) = 1/16) ----
  for (int e = tid; e < 16 * 64; e += 128) {
    int r = e >> 6, c4 = e & 63;
    float4 val = make_float4(0.f, 0.f, 0.f, 0.f);
    if (r < 8) {
      int g = r >> 2, j = r & 3;
      int t = q_start + j;
      if (t > T - 1) t = T - 1;  // reference clamps padded query indices
      const float4* src =
          (const float4*)&q_ws[(size_t)t * (HQ_ * D_) + (h * 2 + g) * D_];
      val = src[c4];
      val.x *= 0.0625f; val.y *= 0.0625f; val.z *= 0.0625f; val.w *= 0.0625f;
    }
    *(float4*)&Qs[r][c4 * 4] = val;
  }
  if (tid < 16) { mS[tid] = -1e30f; lS[tid] = 0.0f; }
  __syncthreads();

  v8f acc0 = {0,0,0,0,0,0,0,0}, acc1 = {0,0,0,0,0,0,0,0};
  v8f acc2 = {0,0,0,0,0,0,0,0}, acc3 = {0,0,0,0,0,0,0,0};
  const int d0 = wave * 64;

  for (int ti = 0; ti < nTiles; ++ti) {
    const int s0 = ti * 16;
    const int cur = ti & 1;
    if (ti < pureTiles) {
      // ---- pipelined TDM path ----
      if (wave == 0) {
        if (ti + 1 < pureTiles) {
          issue_tile(ti + 1);                    // overlap next tile's DMA
          __builtin_amdgcn_s_wait_tensorcnt(2);  // tile ti landed (in-order)
        } else {
          __builtin_amdgcn_s_wait_tensorcnt(0);  // drain pipeline
        }
      }
    } else {
      // ---- boundary/tail tile: per-token staging (cache | new k/v | zero) --
      for (int e = tid; e < 16 * 64; e += 128) {
        int tt = e >> 6, c4 = e & 63;
        int s = s0 + tt;
        float4 kk4 = make_float4(0.f, 0.f, 0.f, 0.f);
        float4 vv4 = make_float4(0.f, 0.f, 0.f, 0.f);
        if (s < hist) {
          int blk = block_offsets[b * MAXBLK_ + (s >> 6)];
          size_t base = (((size_t)blk * BLK_ + (s & 63)) * HK_ + h) * D_;
          kk4 = ((const float4*)&k_cache[base])[c4];
          vv4 = ((const float4*)&v_cache[base])[c4];
        } else if (s < kv_len) {
          int t = q_start + (s - hist);
          size_t base = ((size_t)t * HK_ + h) * D_;
          kk4 = ((const float4*)&k_ws[base])[c4];
          vv4 = ((const float4*)&v_ws[base])[c4];
        }
        *(float4*)&Kd[cur][tt][c4 * 4] = kk4;
        *(float4*)&Vd[cur][tt][c4 * 4] = vv4;
      }
    }
    __syncthreads();

    // ---- wave 0: scores = Qpanel @ K^T  (16x16, contract over D=256) ----
    if (wave == 0) {
      v8f s8 = {0,0,0,0,0,0,0,0};
      for (int kc = 0; kc < D_; kc += 4) {
        v2f a, bb;
        a.x = Qs[ln][kc + 2 * lo];
        a.y = Qs[ln][kc + 2 * lo + 1];
        bb.x = Kd[cur][ln][kc + 2 * lo];  // B[k][n] = K^T[k][n] = Ks[n][k]
        bb.y = Kd[cur][ln][kc + 2 * lo + 1];
        s8 = wmma4(a, bb, s8);
      }
#pragma unroll
      for (int i = 0; i < 8; ++i) {
        int M = i + 8 * lo;
        int spos = s0 + ln;
        bool ok = (M < 8) && (spos <= hist + (M & 3)) && (spos < kv_len);
        Ps[M][ln] = ok ? s8[i] : -1e30f;
      }
    }
    __syncthreads();

    // ---- online softmax, one thread per q row ----
    if (tid < 16) {
      float mOld = mS[tid];
      float tmax = -1e30f;
#pragma unroll
      for (int c = 0; c < 16; ++c) tmax = fmaxf(tmax, Ps[tid][c]);
      float mNew = fmaxf(mOld, tmax);
      float alpha = expf(mOld - mNew);
      float sum = 0.0f;
#pragma unroll
      for (int c = 0; c < 16; ++c) {
        float p = expf(Ps[tid][c] - mNew);
        Ps[tid][c] = p;
        sum += p;
      }
      mS[tid] = mNew;
      lS[tid] = lS[tid] * alpha + sum;
      aS[tid] = alpha;
    }
    __syncthreads();

    // ---- all waves: rescale accumulators, then P @ Vtile over 64-col slab --
    {
      float al[8];
#pragma unroll
      for (int i = 0; i < 8; ++i) al[i] = aS[i + 8 * lo];
#pragma unroll
      for (int i = 0; i < 8; ++i) {
        acc0[i] *= al[i]; acc1[i] *= al[i];
        acc2[i] *= al[i]; acc3[i] *= al[i];
      }
#pragma unroll
      for (int kc = 0; kc < 16; kc += 4) {
        v2f a;
        a.x = Ps[ln][kc + 2 * lo];
        a.y = Ps[ln][kc + 2 * lo + 1];
        v2f b0, b1, b2, b3;
        b0.x = Vd[cur][kc + 2 * lo][d0 + ln];
        b0.y = Vd[cur][kc + 2 * lo + 1][d0 + ln];
        b1.x = Vd[cur][kc + 2 * lo][d0 + 16 + ln];
        b1.y = Vd[cur][kc + 2 * lo + 1][d0 + 16 + ln];
        b2.x = Vd[cur][kc + 2 * lo][d0 + 32 + ln];
        b2.y = Vd[cur][kc + 2 * lo + 1][d0 + 32 + ln];
        b3.x = Vd[cur][kc + 2 * lo][d0 + 48 + ln];
        b3.y = Vd[cur][kc + 2 * lo + 1][d0 + 48 + ln];
        acc0 = wmma4(a, b0, acc0);
        acc1 = wmma4(a, b1, acc1);
        acc2 = wmma4(a, b2, acc2);
        acc3 = wmma4(a, b3, acc3);
      }
    }
    __syncthreads();
  }

  // ---- finalize: divide by l, scatter to flat [T, HQ*D] layout ----
#pragma unroll
  for (int i = 0; i < 8; ++i) {
    int M = i + 8 * lo;
    if (M < 8) {
      int g = M >> 2, j = M & 3;
      if (j < q_len) {
        int t = q_start + j;
        float inv = 1.0f / lS[M];
        size_t row = (size_t)t * (HQ_ * D_) + (size_t)(h * 2 + g) * D_;
        o_ws[row + d0 + ln] = acc0[i] * inv;
        o_ws[row + d0 + 16 + ln] = acc1[i] * inv;
        o_ws[row + d0 + 32 + ln] = acc2[i] * inv;
        o_ws[row + d0 + 48 + ln] = acc3[i] * inv;
      }
    }
  }
}

// ---------------------------------------------------------------------------
extern "C" void kernel_launch(void* const* d_in, const int* in_sizes, int n_in,
                              void* d_out, int out_size, void* d_ws,
                              size_t ws_size, hipStream_t stream) {
  const float* hidden  = (const float*)d_in[0];
  const float* Wq      = (const float*)d_in[1];
  const float* Wk      = (const float*)d_in[2];
  const float* Wv      = (const float*)d_in[3];
  const float* Wo      = (const float*)d_in[4];
  const float* k_cache = (const float*)d_in[5];
  const float* v_cache = (const float*)d_in[6];
  const float* inv_freq  = (const float*)d_in[7];
  const int* pos_ids   = (const int*)d_in[8];
  const int* q_start   = (const int*)d_in[9];
  const int* q_lens    = (const int*)d_in[10];
  const int* kv_lens   = (const int*)d_in[11];
  const int* blk_off   = (const int*)d_in[12];
  float* out = (float*)d_out;

  const int T = in_sizes[0] / HIDDEN_;  // 40

  // workspace layout (floats): q [T,4096] | k [T,2048] | v [T,2048] | o [T,4096]
  float* q_ws = (float*)d_ws;
  float* k_ws = q_ws + (size_t)T * (HQ_ * D_);
  float* v_ws = k_ws + (size_t)T * (HK_ * D_);
  float* o_ws = v_ws + (size_t)T * (HK_ * D_);

  const int mTiles = (T + 15) / 16;
  // QKV projections
  gemm_f32_wmma<<<dim3((HQ_ * D_) / 128, mTiles), 256, 0, stream>>>(
      hidden, Wq, q_ws, T, HQ_ * D_, HIDDEN_);
  gemm_f32_wmma<<<dim3((HK_ * D_) / 128, mTiles), 256, 0, stream>>>(
      hidden, Wk, k_ws, T, HK_ * D_, HIDDEN_);
  gemm_f32_wmma<<<dim3((HK_ * D_) / 128, mTiles), 256, 0, stream>>>(
      hidden, Wv, v_ws, T, HK_ * D_, HIDDEN_);
  // RoPE on q and k
  const int nrope = T * (HQ_ + HK_) * (D_ / 2);
  rope_kernel<<<(nrope + 255) / 256, 256, 0, stream>>>(q_ws, k_ws, pos_ids,
                                                       inv_freq, T);
  // paged attention: one block per (kv head, sequence)
  paged_attn_wmma<<<dim3(HK_, 16), dim3(128), 0, stream>>>(
      q_ws, k_ws, v_ws, k_cache, v_cache, q_start, q_lens, kv_lens, blk_off,
      o_ws, T);
  // output projection
  gemm_f32_wmma<<<dim3(HIDDEN_ / 128, mTiles), 256, 0, stream>>>(
      o_ws, Wo, out, T, HIDDEN_, HQ_ * D_);
}